// FeatureNormalizedMSE_47339129536786
// MI455X (gfx1250) — compile-verified
//
#include <hip/hip_runtime.h>

typedef __attribute__((ext_vector_type(2))) float v2f;
typedef __attribute__((ext_vector_type(8))) float v8f;

#define ROWS   16
#define CHUNK  128
#define LSTR   132          // padded LDS row stride in dwords: bank-rotates rows, keeps 16B align
#define MAXD   2048

// Stage 1: each block reduces 16 rows to one partial sum.
// Row sum-of-squares computed as diag(S * S^T) via V_WMMA_F32_16X16X4_F32,
// exploiting that for this op the A layout and the (B = A^T) layout are identical.
__global__ __launch_bounds__(256) void fnmse_stage1(
    const float* __restrict__ outputs,
    const float* __restrict__ targets,
    const float* __restrict__ weights,
    float* __restrict__ partials,
    int D)
{
    __shared__ float s_lds[ROWS * LSTR];
    __shared__ float w_lds[MAXD];
    __shared__ float diag_lds[8][ROWS];
    __shared__ float row_val[ROWS];
    __shared__ int   cnt_lds[ROWS];

    const int tid  = threadIdx.x;
    const int lane = tid & 31;
    const int wave = tid >> 5;
    const int row  = tid >> 4;     // 16 threads cooperate per row
    const int c16  = tid & 15;

    const long long grow = (long long)blockIdx.x * ROWS + row;
    const float* __restrict__ orow = outputs + grow * (long long)D;
    const float* __restrict__ trow = targets + grow * (long long)D;

    // stage feature weights once (shared by all 16 row-groups every chunk)
    for (int i = tid; i < D; i += 256) w_lds[i] = weights[i];
    if (tid < ROWS) cnt_lds[tid] = 0;
    __syncthreads();

    v8f acc = {};        // acc += S_chunk * S_chunk^T ; diagonal = per-row sum of squares
    int  my_cnt = 0;

    const int a_row = lane & 15;          // WMMA 16x4 f32 A layout
    const int a_k   = (lane >> 4) * 2;    // lanes 0-15: K={0,1}; lanes 16-31: K={2,3}
    const int nchunks = D / CHUNK;

    for (int ch = 0; ch < nchunks; ++ch) {
        const int col0 = ch * CHUNK + c16 * 8;

        // coalesced: 16 threads/row * 32B = 512B contiguous per row
        float4 o0 = *(const float4*)(orow + col0);
        float4 o1 = *(const float4*)(orow + col0 + 4);
        float4 t0 = *(const float4*)(trow + col0);
        float4 t1 = *(const float4*)(trow + col0 + 4);
        float4 w0 = *(const float4*)(&w_lds[col0]);
        float4 w1 = *(const float4*)(&w_lds[col0 + 4]);

        float ov[8] = {o0.x, o0.y, o0.z, o0.w, o1.x, o1.y, o1.z, o1.w};
        float tv[8] = {t0.x, t0.y, t0.z, t0.w, t1.x, t1.y, t1.z, t1.w};
        float wv[8] = {w0.x, w0.y, w0.z, w0.w, w1.x, w1.y, w1.z, w1.w};
        float sv[8];
        #pragma unroll
        for (int j = 0; j < 8; ++j) {
            const bool valid = (tv[j] == tv[j]);        // !isnan(target)
            my_cnt += valid ? 1 : 0;
            const float r = valid ? (tv[j] - ov[j]) : 0.0f;
            sv[j] = r * wv[j];
        }

        __syncthreads();   // previous chunk's WMMA reads of s_lds are done
        float* dst = &s_lds[row * LSTR + c16 * 8];
        *(float4*)(dst)     = make_float4(sv[0], sv[1], sv[2], sv[3]);
        *(float4*)(dst + 4) = make_float4(sv[4], sv[5], sv[6], sv[7]);
        __syncthreads();   // tile visible to all waves

        // each wave consumes its private 16-column slice: 4 K=4 WMMA steps
        const int kbase = wave * 16 + a_k;
        #pragma unroll
        for (int st = 0; st < 4; ++st) {
            v2f a = *(const v2f*)(&s_lds[a_row * LSTR + kbase + st * 4]);
            acc = __builtin_amdgcn_wmma_f32_16x16x4_f32(
                      /*neg_a=*/false, a, /*neg_b=*/false, a,
                      /*c_mod=*/(short)0, acc, /*reuse_a=*/false, /*reuse_b=*/false);
        }
    }

    atomicAdd(&cnt_lds[row], my_cnt);   // deterministic: integer LDS atomic

    // extract diagonal of 16x16 accumulator:
    // M=j lives in (vgpr j, lane j); M=j+8 lives in (vgpr j, lane j+24)
    #pragma unroll
    for (int j = 0; j < 8; ++j) {
        if (lane == j)      diag_lds[wave][j]     = acc[j];
        if (lane == j + 24) diag_lds[wave][j + 8] = acc[j];
    }
    __syncthreads();

    if (tid < ROWS) {
        float ss = 0.0f;
        #pragma unroll
        for (int wv = 0; wv < 8; ++wv) ss += diag_lds[wv][tid];
        row_val[tid] = ss / (float)cnt_lds[tid];
    }
    __syncthreads();

    if (tid == 0) {
        float s = 0.0f;
        #pragma unroll
        for (int r = 0; r < ROWS; ++r) s += row_val[r];
        partials[blockIdx.x] = s;
    }
}

// Stage 2: deterministic fixed-order reduction of per-block partials.
__global__ __launch_bounds__(256) void fnmse_stage2(
    const float* __restrict__ partials, float* __restrict__ out, int n)
{
    __shared__ float red[256];
    const int tid = threadIdx.x;
    float s = 0.0f;
    for (int i = tid; i < n; i += 256) s += partials[i];
    red[tid] = s;
    __syncthreads();
    for (int off = 128; off > 0; off >>= 1) {
        if (tid < off) red[tid] += red[tid + off];
        __syncthreads();
    }
    if (tid == 0) out[0] = red[0];
}

extern "C" void kernel_launch(void* const* d_in, const int* in_sizes, int n_in,
                              void* d_out, int out_size, void* d_ws, size_t ws_size,
                              hipStream_t stream) {
    const float* outputs = (const float*)d_in[0];
    const float* targets = (const float*)d_in[1];
    const float* weights = (const float*)d_in[2];
    float* out = (float*)d_out;

    const int D = in_sizes[2];                       // 2048
    const int N = (int)((long long)in_sizes[0] / D); // 16384
    const int nblocks = N / ROWS;                    // 1024

    float* partials = (float*)d_ws;                  // nblocks * 4B << ws_size

    fnmse_stage1<<<nblocks, 256, 0, stream>>>(outputs, targets, weights, partials, D);
    fnmse_stage2<<<1, 256, 0, stream>>>(partials, out, nblocks);
}